// VTMUpsampler_51221779972153
// MI455X (gfx1250) — compile-verified
//
#include <hip/hip_runtime.h>

typedef __attribute__((ext_vector_type(2))) float v2f;
typedef __attribute__((ext_vector_type(8))) float v8f;

#define TAPS 8

__device__ __forceinline__ void vtm_phase(int i, int s, int& integer, int& frac) {
    // ref = int(i * 16384 / scale); integer = ref >> 4; frac = ref & 15
    unsigned ref = (unsigned)(i * 16384) / (unsigned)s;   // i*16384 < 2^31 for all dims here
    integer = (int)(ref >> 4);
    frac    = (int)(ref & 15u);
}

__device__ __forceinline__ int clampi(int v, int lo, int hi) {
    return v < lo ? lo : (v > hi ? hi : v);
}

// ---------------------------------------------------------------------------
// Pass 1: horizontal 8-tap resample.  x (P,H,W) -> xh (P,H,newW)
// One thread per xh element; lanes run along j for coalesced stores.
// All offsets 32-bit (max byte offset < 2^31) to keep address math light.
// ---------------------------------------------------------------------------
__global__ void __launch_bounds__(256)
vtm_hpass(const float* __restrict__ x, const float* __restrict__ filt,
          const int* __restrict__ scale_p, float* __restrict__ xh,
          int H, int W, int newW) {
    int s = *scale_p;
    int j = blockIdx.x * blockDim.x + threadIdx.x;
    int h = blockIdx.y;
    int p = blockIdx.z;
    if (j >= newW) return;

    int integer, frac;
    vtm_phase(j, s, integer, frac);
    const float* row = x + (unsigned)((p * H + h) * W);   // scalar base
    const float* cw  = filt + (unsigned)(frac * TAPS);

    float acc = 0.f;
#pragma unroll
    for (int k = 0; k < TAPS; ++k) {
        int idx = clampi(integer + k - (TAPS / 2) + 1, 0, W - 1);
        acc += cw[k] * row[(unsigned)idx];
    }
    xh[(unsigned)((p * H + h) * newW) + (unsigned)j] = acc;
}

// ---------------------------------------------------------------------------
// Pass 2: vertical 8-tap resample + /4096.  xh (P,H,newW) -> out (P,newH,newW)
// One wave32 per 16x16 output tile.  If all 16 rows of the tile share one
// clamped tap-row vector (scalar runtime check via readlane), the tile is
// D = A(16x8) x B(8x16) via two chained V_WMMA_F32_16X16X4_F32.
// Tile id forced to SGPRs with readfirstlane so the branch is scalar and
// EXEC is structurally all-1s at the WMMA ops.
// ---------------------------------------------------------------------------
__global__ void __launch_bounds__(256)
vtm_vpass(const float* __restrict__ xh, const float* __restrict__ filt,
          const int* __restrict__ scale_p, float* __restrict__ out,
          int H, int newH, int newW, int tilesY, int tilesX, int totalTiles) {
    int s = *scale_p;
    int wave = blockIdx.x * (blockDim.x >> 5) + (threadIdx.x >> 5);
    wave = __builtin_amdgcn_readfirstlane(wave);        // wave-uniform -> SGPR
    if (wave >= totalTiles) return;                     // scalar exit
    int lane = threadIdx.x & 31;

    int tilesPerPlane = tilesY * tilesX;
    int p   = wave / tilesPerPlane;                     // all scalar math
    int rem = wave - p * tilesPerPlane;
    int ty  = rem / tilesX;
    int tx  = rem - ty * tilesX;
    int Y0 = ty * 16, X0 = tx * 16;

    int m    = lane & 15;    // output row within tile (A: M = lane&15)
    int hsel = lane >> 4;    // K half select for A/B operand layout

    // Per-row taps/coefs for row Y0+m (lanes 16..31 mirror lanes 0..15)
    int integer, frac;
    vtm_phase(Y0 + m, s, integer, frac);
    const float* chp = filt + (unsigned)(frac * TAPS);

    int   r[TAPS];
    float c[TAPS];
#pragma unroll
    for (int k = 0; k < TAPS; ++k) {
        r[k] = clampi(integer + k - (TAPS / 2) + 1, 0, H - 1);
        c[k] = chp[k];
    }

    // Uniformity: tap indices are monotonic in m, so endpoint equality
    // (row m=0 vs m=15) implies all 16 rows share the tap-row vector.
    // readlane makes rs[] SGPRs and the condition a scalar branch.
    int  rs[TAPS];
    bool uni = true;
#pragma unroll
    for (int k = 0; k < TAPS; ++k) {
        rs[k]  = __builtin_amdgcn_readlane(r[k], 0);
        int re = __builtin_amdgcn_readlane(r[k], 15);
        uni = uni && (rs[k] == re);
    }

    const float* plane  = xh  + (unsigned)(p * H    * newW);   // scalar base
    float*       oplane = out + (unsigned)(p * newH * newW);   // scalar base
    const float  norm = 1.0f / 4096.0f;
    int n = m;                                          // B/D column = lane&15

    if (uni && (Y0 + 16 <= newH)) {
        // ---- WMMA path: D(16x16) = A(16x8) * B(8x16), K split 4+4 ----
        // A layout (f32 16x4): lanes 0-15 M=0-15 {K=0,1}; lanes 16-31 {K=2,3}
        v2f aA, aB;
        aA.x = hsel ? c[2] : c[0];  aA.y = hsel ? c[3] : c[1];
        aB.x = hsel ? c[6] : c[4];  aB.y = hsel ? c[7] : c[5];

        // B layout (f32 4x16): lanes 0-15 N=0-15 {K=0,1}; lanes 16-31 {K=2,3}
        int rb0 = hsel ? rs[2] : rs[0];
        int rb1 = hsel ? rs[3] : rs[1];
        int rb2 = hsel ? rs[6] : rs[4];
        int rb3 = hsel ? rs[7] : rs[5];
        unsigned colOff = (unsigned)(X0 + n);
        v2f bA, bB;
        bA.x = plane[(unsigned)(rb0 * newW) + colOff];
        bA.y = plane[(unsigned)(rb1 * newW) + colOff];
        bB.x = plane[(unsigned)(rb2 * newW) + colOff];
        bB.y = plane[(unsigned)(rb3 * newW) + colOff];

        v8f acc = {};
        acc = __builtin_amdgcn_wmma_f32_16x16x4_f32(false, aA, false, bA,
                                                    (short)0, acc, false, false);
        acc = __builtin_amdgcn_wmma_f32_16x16x4_f32(false, aB, false, bB,
                                                    (short)0, acc, false, false);

        // D layout: vgpr i -> row Y0 + i + 8*hsel, col X0 + (lane&15)
        unsigned base = (unsigned)((Y0 + hsel * 8) * newW) + colOff;
#pragma unroll
        for (int i = 0; i < 8; ++i) {
            oplane[base + (unsigned)(i * newW)] = acc[i] * norm;  // imm-offset stores
        }
    } else {
        // ---- VALU fallback: edge tiles / non-uniform tap rows ----
#pragma unroll
        for (int i = 0; i < 8; ++i) {
            int row = Y0 + hsel * 8 + i;
            if (row < newH) {
                int ig, fr;
                vtm_phase(row, s, ig, fr);
                const float* cc = filt + (unsigned)(fr * TAPS);
                float acc = 0.f;
#pragma unroll
                for (int k = 0; k < TAPS; ++k) {
                    int rr = clampi(ig + k - (TAPS / 2) + 1, 0, H - 1);
                    acc += cc[k] * plane[(unsigned)(rr * newW + X0 + n)];
                }
                oplane[(unsigned)(row * newW + X0 + n)] = acc * norm;
            }
        }
    }
}

// ---------------------------------------------------------------------------
// Fallback if workspace is too small: fully fused direct 8x8-tap kernel.
// ---------------------------------------------------------------------------
__global__ void __launch_bounds__(256)
vtm_fused(const float* __restrict__ x, const float* __restrict__ filt,
          const int* __restrict__ scale_p, float* __restrict__ out,
          int H, int W, int newH, int newW) {
    int s = *scale_p;
    int xo = blockIdx.x * blockDim.x + threadIdx.x;
    int yo = blockIdx.y;
    int p  = blockIdx.z;
    if (xo >= newW) return;

    int iy, fy, ix, fx;
    vtm_phase(yo, s, iy, fy);
    vtm_phase(xo, s, ix, fx);
    const float* chv = filt + (unsigned)(fy * TAPS);
    const float* cwv = filt + (unsigned)(fx * TAPS);
    const float* plane = x + (unsigned)(p * H * W);

    float acc = 0.f;
#pragma unroll
    for (int ky = 0; ky < TAPS; ++ky) {
        int rr = clampi(iy + ky - (TAPS / 2) + 1, 0, H - 1);
        const float* row = plane + (unsigned)(rr * W);
        float hacc = 0.f;
#pragma unroll
        for (int kx = 0; kx < TAPS; ++kx) {
            int cidx = clampi(ix + kx - (TAPS / 2) + 1, 0, W - 1);
            hacc += cwv[kx] * row[(unsigned)cidx];
        }
        acc += chv[ky] * hacc;
    }
    out[(unsigned)((p * newH + yo) * newW) + (unsigned)xo] = acc * (1.0f / 4096.0f);
}

extern "C" void kernel_launch(void* const* d_in, const int* in_sizes, int n_in,
                              void* d_out, int out_size, void* d_ws, size_t ws_size,
                              hipStream_t stream) {
    (void)in_sizes; (void)n_in; (void)out_size;
    const float* x     = (const float*)d_in[0];
    const float* filt  = (const float*)d_in[1];
    const int*   scale = (const int*)d_in[2];
    float*       out   = (float*)d_out;

    const int B = 2, C = 8, H = 540, W = 960, S = 3;
    const int P = B * C;
    const int newW = W * S;   // 2880
    const int newH = H * S;   // 1620

    size_t wsNeed = (size_t)P * H * newW * sizeof(float);  // 99.5 MB (L2-resident)
    if (ws_size >= wsNeed) {
        float* xh = (float*)d_ws;
        dim3 g1((newW + 255) / 256, H, P);
        vtm_hpass<<<g1, 256, 0, stream>>>(x, filt, scale, xh, H, W, newW);

        int tilesY = (newH + 15) / 16;          // 102
        int tilesX = newW / 16;                 // 180 (exact)
        int totalTiles = tilesY * tilesX * P;   // 293760
        int wavesPerBlock = 256 / 32;
        int blocks = (totalTiles + wavesPerBlock - 1) / wavesPerBlock;
        vtm_vpass<<<blocks, 256, 0, stream>>>(xh, filt, scale, out,
                                              H, newH, newW, tilesY, tilesX, totalTiles);
    } else {
        dim3 g((newW + 255) / 256, newH, P);
        vtm_fused<<<g, 256, 0, stream>>>(x, filt, scale, out, H, W, newH, newW);
    }
}